// LieRE_4896262717990
// MI455X (gfx1250) — compile-verified
//
#include <hip/hip_runtime.h>
#include <hip/hip_bf16.h>

typedef __attribute__((ext_vector_type(2))) float v2f;
typedef __attribute__((ext_vector_type(8))) float v8f;

// LDS row stride for 64x64 tiles: 68 floats. 68 mod 64 = 4, so walking down a
// column strides banks by 4 -> 16 rows hit 16 distinct banks (conflict-free
// for the WMMA fragment gathers).
#define LSTR 68

__device__ __forceinline__ v8f v8f_zero() {
  v8f z = {0.f, 0.f, 0.f, 0.f, 0.f, 0.f, 0.f, 0.f};
  return z;
}

// One wave computes one 16x16 tile of C = A(64x64) @ B(64x64), A/B in LDS
// (row-major, stride LSTR). fp32 WMMA 16x16x4, K swept in 16 steps.
// ISA 7.12.2 fp32 layouts:
//   A 16x4 : lanes 0-15 & 16-31 both carry M=0..15; VGPR0=K{0|2}, VGPR1=K{1|3}
//   B 4x16 : VGPR r carries K=r (lanes 0-15) and K=r+2 (lanes 16-31), N=lane&15
//   C/D    : VGPR r carries row m0+r (lanes 0-15) / m0+r+8 (lanes 16-31)
__device__ __forceinline__ v8f mm_tile64(const float* A, const float* B,
                                         int m0, int n0, int n, int hi) {
  v8f c = v8f_zero();
#pragma unroll
  for (int k0 = 0; k0 < 64; k0 += 4) {
    v2f a, b;
    a.x = A[(m0 + n) * LSTR + k0 + 2 * hi];
    a.y = A[(m0 + n) * LSTR + k0 + 2 * hi + 1];
    b.x = B[(k0 + 2 * hi) * LSTR + n0 + n];
    b.y = B[(k0 + 2 * hi + 1) * LSTR + n0 + n];
    c = __builtin_amdgcn_wmma_f32_16x16x4_f32(false, a, false, b,
                                              (short)0, c, false, false);
  }
  return c;
}

// Full 64x64 @ 64x64 -> 64x64 in LDS with 16 waves (512 threads).
__device__ __forceinline__ void mm64(const float* A, const float* B, float* C,
                                     int tid) {
  const int lane = tid & 31;
  const int wave = tid >> 5;          // 0..15
  const int n = lane & 15;
  const int hi = lane >> 4;
  const int m0 = (wave >> 2) << 4;    // 4 M tiles
  const int n0 = (wave & 3) << 4;     // 4 N tiles
  v8f c = mm_tile64(A, B, m0, n0, n, hi);
#pragma unroll
  for (int r = 0; r < 8; ++r)
    C[(m0 + r + 8 * hi) * LSTR + n0 + n] = c[r];
}

// ---------------------------------------------------------------------------
// Kernel 1: per spatial position (y,x), build the 64x64 generator
//   G = y*skew0 + x*skew1  (skew[i,j] = emb[i,j]-emb[j,i], -0.5 cancels)
// and compute rot = expm(G) via scaling-and-squaring + order-8 Horner Taylor.
// One block (512 thr / 16 waves) per position; result -> rot[pos][64][64].
// ---------------------------------------------------------------------------
__global__ __launch_bounds__(512) void lie_expm_kernel(
    const float* __restrict__ emb, float* __restrict__ rot) {
  __shared__ float X[64 * LSTR];
  __shared__ float P[64 * LSTR];
  __shared__ float Q[64 * LSTR];
  __shared__ float rn[64];
  __shared__ int sh_s;

  const int tid = threadIdx.x;
  const int pos = blockIdx.x;          // y*32 + x
  const int y = pos >> 5, x = pos & 31;
  // H==W==32 -> xlim = ylim = 1; linspace(-1, 1, 32)
  const float py = -1.0f + 2.0f * (float)y * (1.0f / 31.0f);
  const float px = -1.0f + 2.0f * (float)x * (1.0f / 31.0f);

  // Generator into X
#pragma unroll
  for (int e = 0; e < 8; ++e) {
    const int idx = tid * 8 + e;
    const int i = idx >> 6, j = idx & 63;
    float s0, s1;
    if (i < j) {
      s0 = emb[i * 64 + j];
      s1 = emb[4096 + i * 64 + j];
    } else if (i > j) {
      s0 = -emb[j * 64 + i];
      s1 = -emb[4096 + j * 64 + i];
    } else {
      s0 = 0.f;
      s1 = 0.f;
    }
    X[i * LSTR + j] = py * s0 + px * s1;
  }
  __syncthreads();

  // inf-norm (max abs row sum) -> scaling exponent s so that ||X/2^s|| <= 0.25
  if (tid < 64) {
    float s = 0.f;
    for (int j = 0; j < 64; ++j) s += fabsf(X[tid * LSTR + j]);
    rn[tid] = s;
  }
  __syncthreads();
  if (tid == 0) {
    float m = 0.f;
    for (int i = 0; i < 64; ++i) m = fmaxf(m, rn[i]);
    int s = 0;
    if (m > 0.25f) s = (int)ceilf(log2f(m * 4.0f));
    sh_s = min(max(s, 0), 30);
  }
  __syncthreads();
  const int sc = sh_s;
  const float f = exp2f((float)(-sc));
#pragma unroll
  for (int e = 0; e < 8; ++e) {
    const int idx = tid * 8 + e;
    X[(idx >> 6) * LSTR + (idx & 63)] *= f;
  }
  __syncthreads();

  // Horner Taylor, order 8: P = I + X/8; for k=7..1: P = I + (X @ P)/k
  // Fully unrolled so 1/k folds to an inline fp literal (no runtime divide
  // polluting the WMMA stream).
#pragma unroll
  for (int e = 0; e < 8; ++e) {
    const int idx = tid * 8 + e;
    const int i = idx >> 6, j = idx & 63;
    P[i * LSTR + j] = X[i * LSTR + j] * 0.125f + (i == j ? 1.f : 0.f);
  }
  __syncthreads();
#pragma unroll
  for (int k = 7; k >= 1; --k) {
    mm64(X, P, Q, tid);
    __syncthreads();
    const float inv = 1.0f / (float)k;   // compile-time constant after unroll
#pragma unroll
    for (int e = 0; e < 8; ++e) {
      const int idx = tid * 8 + e;
      const int i = idx >> 6, j = idx & 63;
      P[i * LSTR + j] = Q[i * LSTR + j] * inv + (i == j ? 1.f : 0.f);
    }
    __syncthreads();
  }

  // Repeated squaring: rot = P^(2^sc)
  float* cur = P;
  float* oth = Q;
  for (int it = 0; it < sc; ++it) {
    mm64(cur, cur, oth, tid);
    __syncthreads();
    float* t = cur;
    cur = oth;
    oth = t;
  }

  float* dst = rot + pos * 4096;
#pragma unroll
  for (int e = 0; e < 8; ++e) {
    const int idx = tid * 8 + e;
    dst[idx] = cur[(idx >> 6) * LSTR + (idx & 63)];
  }
}

// ---------------------------------------------------------------------------
// Kernel 2: apply. Per position (y,x): out_slab[512,64] = in_slab[512,64] @ rot
// where slab row r = (n, q) = (r>>4, r&15), element = input[n][y][x][q][:].
// One block (256 thr / 8 waves) per position; rot staged in LDS; each wave
// sweeps 4 M-tiles x 4 N-tiles, reusing the A fragment across the 4 N WMMAs.
// Each global input element is read exactly once per block -> memory-floor
// traffic (134 MB in + 134 MB out ~= 11.6 us at 23.3 TB/s).
// ---------------------------------------------------------------------------
__global__ __launch_bounds__(256) void lie_apply_kernel(
    const float* __restrict__ in, const float* __restrict__ rot,
    float* __restrict__ out) {
  __shared__ float R[64 * LSTR];

  const int tid = threadIdx.x;
  const int pos = blockIdx.x;          // y*32 + x
  const int posoff = pos * 1024;       // (y*W + x) * h * d
  constexpr int NSTRIDE = 32 * 32 * 16 * 64;  // per-image stride = 1048576

  const float* src = rot + pos * 4096;
#pragma unroll
  for (int e = 0; e < 16; ++e) {
    const int idx = tid + e * 256;
    R[(idx >> 6) * LSTR + (idx & 63)] = src[idx];
  }
  __syncthreads();

  const int lane = tid & 31;
  const int wave = tid >> 5;           // 0..7
  const int n = lane & 15;
  const int hi = lane >> 4;

  for (int mt = wave; mt < 32; mt += 8) {
    const int m0 = mt << 4;
    const int mrowA = m0 + n;          // A-fragment row for this lane
    const long baseA =
        (long)(mrowA >> 4) * NSTRIDE + posoff + (mrowA & 15) * 64;

    v8f acc[4];
#pragma unroll
    for (int nt = 0; nt < 4; ++nt) acc[nt] = v8f_zero();

#pragma unroll
    for (int k0 = 0; k0 < 64; k0 += 4) {
      const float2 av =
          *reinterpret_cast<const float2*>(in + baseA + k0 + 2 * hi);
      v2f a;
      a.x = av.x;
      a.y = av.y;
#pragma unroll
      for (int nt = 0; nt < 4; ++nt) {
        v2f b;
        b.x = R[(k0 + 2 * hi) * LSTR + nt * 16 + n];
        b.y = R[(k0 + 2 * hi + 1) * LSTR + nt * 16 + n];
        acc[nt] = __builtin_amdgcn_wmma_f32_16x16x4_f32(
            false, a, false, b, (short)0, acc[nt], false, false);
      }
    }

#pragma unroll
    for (int nt = 0; nt < 4; ++nt) {
#pragma unroll
      for (int r = 0; r < 8; ++r) {
        const int mrow = m0 + r + 8 * hi;
        out[(long)(mrow >> 4) * NSTRIDE + posoff + (mrow & 15) * 64 +
            nt * 16 + n] = acc[nt][r];
      }
    }
  }
}

extern "C" void kernel_launch(void* const* d_in, const int* in_sizes, int n_in,
                              void* d_out, int out_size, void* d_ws,
                              size_t ws_size, hipStream_t stream) {
  const float* in = (const float*)d_in[0];   // [32,32,32,16,64] f32
  const float* emb = (const float*)d_in[1];  // [2,64,64] f32
  float* out = (float*)d_out;                // [32,32,32,16,64] f32
  float* rot = (float*)d_ws;                 // [1024,64,64] f32 = 16.8 MB

  lie_expm_kernel<<<1024, 512, 0, stream>>>(emb, rot);
  lie_apply_kernel<<<1024, 256, 0, stream>>>(in, rot, out);
}